// RecurrentFullAttention_73976516706537
// MI455X (gfx1250) — compile-verified
//
#include <hip/hip_runtime.h>

// Recurrent full-attention decode step, MI455X (gfx1250), wave32.
// N=16 H=32 S=4096 E=D=128. One workgroup per (n,h) head; 512 WGs, 256 thr (8 waves).
// Bandwidth-bound (~4.3 GB @ 23.3 TB/s => ~185us floor): every K/V row is read once
// (serving both the cache copy and the attention math) and written once (NT hints).
// Rows are staged in a 16-register batch so 16 global_load_b128 are in flight per
// wave (vs. the load->waitcnt0->store serialization the naive loop produced).

typedef __attribute__((ext_vector_type(2))) float v2f;
typedef __attribute__((ext_vector_type(4))) float v4f;
typedef __attribute__((ext_vector_type(8))) float v8f;

#define N_      16
#define H_      32
#define S_      4096
#define E_      128
#define D_      128
#define NH      (N_ * H_)          // 512 heads
#define BLOCK   256
#define NWAVES  8
#define NTILES  (S_ / 16)          // 256 tiles of 16 rows
#define TPAD    132                // padded LDS tile row (floats): bank-conflict-free gathers

static __device__ __forceinline__ v4f ntload4(const float* p) {
    return __builtin_nontemporal_load((const v4f*)p);
}
static __device__ __forceinline__ void ntstore4(float* p, v4f v) {
    __builtin_nontemporal_store(v, (v4f*)p);
}

__launch_bounds__(BLOCK)
__global__ void rfa_decode_kernel(const float* __restrict__ q,
                                  const float* __restrict__ knew,
                                  const float* __restrict__ vnew,
                                  const float* __restrict__ mk,
                                  const float* __restrict__ mv,
                                  float* __restrict__ out_v,
                                  float* __restrict__ out_k,
                                  float* __restrict__ out_val)
{
    extern __shared__ float smem[];
    float* qbuf   = smem;                         // 128 floats
    float* scores = qbuf + E_;                    // S_+8 floats (scores then exp weights)
    float* tiles  = scores + (S_ + 8);            // 8 waves * 16 * TPAD floats
    float* red    = tiles + NWAVES * 16 * TPAD;   // 1024 floats (reductions)

    const int head = blockIdx.x;
    const int t    = threadIdx.x;
    const int wave = t >> 5;
    const int lane = t & 31;
    const float temp = 0.08838834764831845f;      // 1/sqrt(E)

    // ---- stage q, seed score slot for the appended key ----
    if (t < E_) qbuf[t] = q[(size_t)head * E_ + t];
    if (t == BLOCK - 1) scores[S_] = 0.0f;
    __syncthreads();

    if (t < E_) {
        const float kn = knew[(size_t)head * E_ + t];
        out_k[((size_t)head * (S_ + 1) + S_) * E_ + t] = kn;   // append key row s=S
        atomicAdd(&scores[S_], qbuf[t] * kn * temp);           // ds_add_f32
    }

    // ---- Phase A: stream K tiles -> out_keys, score via V_WMMA_F32_16X16X4_F32 ----
    // A layout (16x4 fp32): lanes 0..15 hold {K=0,K=1}, lanes 16..31 hold {K=2,K=3},
    // M = lane&15.  B = q chunk broadcast over all 16 columns (same half-wave K split),
    // so C[i][j] == score(row i) for every j; rows 0..7 on lanes 0..15, 8..15 on 16..31.
    float* tile = tiles + wave * 16 * TPAD;
    const int rowL  = lane & 15;
    const int khalf = (lane >> 4) << 1;           // 0 or 2

    for (int tileId = wave; tileId < NTILES; tileId += NWAVES) {
        const int row0 = tileId << 4;
        const float* src = mk   + ((size_t)head * S_       + row0) * E_ + 4 * lane;
        float*       dst = out_k + ((size_t)head * (S_ + 1) + row0) * E_ + 4 * lane;

        v4f buf[16];
#pragma unroll
        for (int r = 0; r < 16; ++r)              // 16 loads in flight (MLP)
            buf[r] = ntload4(src + (size_t)r * E_);
#pragma unroll
        for (int r = 0; r < 16; ++r) {            // drain: copy out + stage in LDS
            ntstore4(dst + (size_t)r * E_, buf[r]);
            *(v4f*)&tile[r * TPAD + 4 * lane] = buf[r];
        }

        v8f acc = {0.f, 0.f, 0.f, 0.f, 0.f, 0.f, 0.f, 0.f};
        const float* trow = &tile[rowL * TPAD];
#pragma unroll
        for (int c = 0; c < 32; ++c) {            // E=128 in K-chunks of 4
            const int kb = 4 * c + khalf;
            const v2f a = *(const v2f*)&trow[kb];  // conflict-free ds_load_b64
            const v2f b = *(const v2f*)&qbuf[kb];  // broadcast ds_load_b64
            acc = __builtin_amdgcn_wmma_f32_16x16x4_f32(
                false, a, false, b, (short)0, acc, false, false);
        }
        if (rowL == 0) {                          // lane 0 -> rows 0..7, lane 16 -> 8..15
            const int rbase = row0 + ((lane >> 4) << 3);
#pragma unroll
            for (int r = 0; r < 8; ++r) scores[rbase + r] = acc[r] * temp;
        }
    }
    __syncthreads();

    // ---- Phase B: softmax stats over S_+1 scores (normalize at the very end) ----
    float m = -INFINITY;
    for (int i = t; i <= S_; i += BLOCK) m = fmaxf(m, scores[i]);
    red[t] = m; __syncthreads();
    for (int off = BLOCK / 2; off > 0; off >>= 1) {
        if (t < off) red[t] = fmaxf(red[t], red[t + off]);
        __syncthreads();
    }
    const float smax = red[0];
    __syncthreads();

    float lsum = 0.f;
    for (int i = t; i <= S_; i += BLOCK) {
        const float e = __expf(scores[i] - smax);
        scores[i] = e;                            // keep unnormalized weights
        lsum += e;
    }
    red[t] = lsum; __syncthreads();
    for (int off = BLOCK / 2; off > 0; off >>= 1) {
        if (t < off) red[t] += red[t + off];
        __syncthreads();
    }
    const float invSum = 1.0f / red[0];
    __syncthreads();                              // red reused below

    // ---- Phase C: stream V tiles -> out_values, accumulate sum_s e[s]*V[s,d] ----
    v4f acc = {0.f, 0.f, 0.f, 0.f};               // lane covers cols 4*lane..4*lane+3
    for (int tileId = wave; tileId < NTILES; tileId += NWAVES) {
        const int row0 = tileId << 4;
        const float* src = mv     + ((size_t)head * S_       + row0) * D_ + 4 * lane;
        float*       dst = out_val + ((size_t)head * (S_ + 1) + row0) * D_ + 4 * lane;

        v4f buf[16];
#pragma unroll
        for (int r = 0; r < 16; ++r)              // 16 loads in flight (MLP)
            buf[r] = ntload4(src + (size_t)r * D_);
#pragma unroll
        for (int r = 0; r < 16; ++r) {
            ntstore4(dst + (size_t)r * D_, buf[r]);
            acc = acc + buf[r] * scores[row0 + r];
        }
    }
    if (wave == 0) {                              // appended value row s=S
        const v4f vv = ntload4(vnew + (size_t)head * D_ + 4 * lane);
        ntstore4(out_val + ((size_t)head * (S_ + 1) + S_) * D_ + 4 * lane, vv);
        acc = acc + vv * scores[S_];
    }
    *(v4f*)&red[wave * D_ + 4 * lane] = acc;      // cross-wave reduction
    __syncthreads();
    if (t < D_) {
        float s = 0.f;
#pragma unroll
        for (int w = 0; w < NWAVES; ++w) s += red[w * D_ + t];
        out_v[(size_t)head * D_ + t] = s * invSum;
    }
}

extern "C" void kernel_launch(void* const* d_in, const int* in_sizes, int n_in,
                              void* d_out, int out_size, void* d_ws, size_t ws_size,
                              hipStream_t stream) {
    (void)in_sizes; (void)n_in; (void)out_size; (void)d_ws; (void)ws_size;
    const float* q  = (const float*)d_in[0];
    const float* k  = (const float*)d_in[1];
    const float* v  = (const float*)d_in[2];
    const float* mk = (const float*)d_in[3];
    const float* mv = (const float*)d_in[4];

    float* out      = (float*)d_out;               // [v | keys | values], flat
    float* out_v    = out;
    float* out_k    = out_v + (size_t)NH * D_;
    float* out_val  = out_k + (size_t)NH * (S_ + 1) * (size_t)E_;

    const size_t smem_bytes =
        (size_t)(E_ + (S_ + 8) + NWAVES * 16 * TPAD + 1024) * sizeof(float);

    rfa_decode_kernel<<<dim3(NH), dim3(BLOCK), smem_bytes, stream>>>(
        q, k, v, mk, mv, out_v, out_k, out_val);
}